// MaskAttnNet_56977036149409
// MI455X (gfx1250) — compile-verified
//
#include <hip/hip_runtime.h>
#include <hip/hip_bf16.h>

typedef __attribute__((ext_vector_type(2))) float v2f;
typedef __attribute__((ext_vector_type(8))) float v8f;

#define L_DIM   12
#define B_DIM   4
#define H_DIM   12
#define S_DIM   785
#define NP4     784          // 49 * 16
#define K_SCOPE 588
#define NPART   12           // one layer per partition
#define LN_EPS  1e-5f

// ---------------------------------------------------------------------------
// Kernel 1: column sums of att_maps[l,b,h,1:,1:] via V_WMMA_F32_16X16X4_F32.
// A = all-ones 16x4, B = 4x16 tile (4 consecutive rows x 16 consecutive cols),
// C accumulates running column sums in f32.  Every row of D equals the column
// sums, so lane L (L<16) reads colsum[L] from c[0].
// Grid: x = 49 column tiles, y = layer (12), z = batch (4).  256 thr = 8 waves.
// Each block writes its own slice of partials[NPART][B][784] (no atomics).
// ---------------------------------------------------------------------------
__global__ void __launch_bounds__(256)
maskattn_colsum_wmma(const float* __restrict__ att, float* __restrict__ partials)
{
    const int jt    = blockIdx.x;           // 0..48
    const int part  = blockIdx.y;           // layer index 0..11
    const int b     = blockIdx.z;           // 0..3
    const int tid   = threadIdx.x;
    const int wave  = tid >> 5;
    const int lane  = tid & 31;
    const int laneN = lane & 15;
    const int rowOff = (lane >> 4) << 1;    // lanes 0-15 -> rows +0/+1, lanes 16-31 -> +2/+3

    const size_t colIdx = (size_t)(1 + jt * 16 + laneN);   // skip CLS column

    v8f c = {};
    v2f av; av.x = 1.0f; av.y = 1.0f;       // all-ones A matrix

    for (int h = 0; h < H_DIM; ++h) {
        const float* base =
            att + (((size_t)part * B_DIM + b) * H_DIM + h) * (size_t)S_DIM * S_DIM;
        // 196 chunks of 4 rows cover source rows 1..784 (skip CLS row)
        for (int ic = wave; ic < 196; ic += 8) {
            const int r = 1 + 4 * ic + rowOff;
            const float* p0 = base + (size_t)r * S_DIM + colIdx;
            v2f bv;
            bv.x = p0[0];
            bv.y = p0[S_DIM];
            c = __builtin_amdgcn_wmma_f32_16x16x4_f32(
                    false, av, false, bv, (short)0, c, false, false);
        }
    }

    __shared__ float acc[8][16];
    if (lane < 16) acc[wave][laneN] = c[0];
    __syncthreads();
    if (tid < 16) {
        float t = 0.0f;
#pragma unroll
        for (int w = 0; w < 8; ++w) t += acc[w][tid];
        partials[((size_t)part * B_DIM + b) * NP4 + jt * 16 + tid] = t;
    }
}

// ---------------------------------------------------------------------------
// Kernel 2: per-batch epilogue.  Reduce partials -> s (scaled by 1/(L*H)),
// LayerNorm (two-pass mean/var), sigmoid, exact k-th smallest via 1024-wide
// bitonic sort in LDS (pad +INF), then threshold mask.
// Grid: 4 blocks (one per batch row), 1024 threads.
// ---------------------------------------------------------------------------
__global__ void __launch_bounds__(1024)
maskattn_finalize(const float* __restrict__ partials,
                  const float* __restrict__ gamma,
                  const float* __restrict__ beta,
                  float* __restrict__ out)
{
    const int b   = blockIdx.x;
    const int tid = threadIdx.x;            // 0..1023

    __shared__ float porig[NP4];
    __shared__ float sortv[1024];
    __shared__ float red[1024];

    float s = 0.0f;
    if (tid < NP4) {
#pragma unroll
        for (int p = 0; p < NPART; ++p)
            s += partials[((size_t)p * B_DIM + b) * NP4 + tid];
        s *= (1.0f / (float)(L_DIM * H_DIM));
    }

    // mean
    red[tid] = (tid < NP4) ? s : 0.0f;
    __syncthreads();
    for (int off = 512; off > 0; off >>= 1) {
        if (tid < off) red[tid] += red[tid + off];
        __syncthreads();
    }
    const float mu = red[0] * (1.0f / (float)NP4);
    __syncthreads();

    // variance
    float d = (tid < NP4) ? (s - mu) : 0.0f;
    red[tid] = d * d;
    __syncthreads();
    for (int off = 512; off > 0; off >>= 1) {
        if (tid < off) red[tid] += red[tid + off];
        __syncthreads();
    }
    const float var = red[0] * (1.0f / (float)NP4);
    __syncthreads();

    // LayerNorm + sigmoid
    float pv = __builtin_inff();
    if (tid < NP4) {
        const float inv = 1.0f / sqrtf(var + LN_EPS);
        const float ln  = (s - mu) * inv * gamma[tid] + beta[tid];
        pv = 1.0f / (1.0f + expf(-ln));
        porig[tid] = pv;
    }
    sortv[tid] = pv;                        // +INF padding sorts to the top
    __syncthreads();

    // bitonic sort, ascending, 1024 elements
    for (int k = 2; k <= 1024; k <<= 1) {
        for (int j2 = k >> 1; j2 > 0; j2 >>= 1) {
            const int ixj = tid ^ j2;
            if (ixj > tid) {
                const float a0 = sortv[tid];
                const float a1 = sortv[ixj];
                const bool up = ((tid & k) == 0);
                if ((a0 > a1) == up) {
                    sortv[tid] = a1;
                    sortv[ixj] = a0;
                }
            }
            __syncthreads();
        }
    }

    const float thr = sortv[K_SCOPE - 1];
    if (tid < NP4)
        out[(size_t)b * NP4 + tid] = (porig[tid] > thr) ? 1.0f : 0.0f;
}

// ---------------------------------------------------------------------------
extern "C" void kernel_launch(void* const* d_in, const int* in_sizes, int n_in,
                              void* d_out, int out_size, void* d_ws, size_t ws_size,
                              hipStream_t stream)
{
    (void)in_sizes; (void)n_in; (void)out_size; (void)ws_size;
    const float* att   = (const float*)d_in[0];   // [12,4,12,785,785] f32
    const float* gamma = (const float*)d_in[1];   // [784] f32
    const float* beta  = (const float*)d_in[2];   // [784] f32
    float* out         = (float*)d_out;           // [4,784] f32 mask
    float* partials    = (float*)d_ws;            // [12][4][784] f32 = 147 KB

    dim3 g1(NP4 / 16, NPART, B_DIM);              // (49, 12, 4)
    maskattn_colsum_wmma<<<g1, 256, 0, stream>>>(att, partials);
    maskattn_finalize<<<B_DIM, 1024, 0, stream>>>(partials, gamma, beta, out);
}